// fastGTN_45019847197465
// MI455X (gfx1250) — compile-verified
//
#include <hip/hip_runtime.h>
#include <hip/hip_bf16.h>
#include <math.h>
#include <stdint.h>

typedef __attribute__((ext_vector_type(2))) float v2f;
typedef __attribute__((ext_vector_type(8))) float v8f;

#define LEAKY 0.2f

// float atomic max via signed/unsigned integer ordering trick (valid with -inf init)
__device__ __forceinline__ void atomicMaxF(float* addr, float v) {
  if (v >= 0.0f) atomicMax((int*)addr, __float_as_int(v));
  else           atomicMin((unsigned int*)addr, __float_as_uint(v));
}

__global__ void fill_kernel(float* __restrict__ p, float v, long n) {
  long i = (long)blockIdx.x * blockDim.x + threadIdx.x;
  if (i < n) p[i] = v;
}

// ---------------------------------------------------------------------------
// Out[N x 64] = A[N x 64] * B[64 x 64], fp32 via V_WMMA_F32_16X16X4_F32.
// One wave computes a 16-row x 64-col tile. A register-blocked (loaded once,
// reused for all 4 column tiles). B staged into LDS with the CDNA5 async
// global->LDS copy path (GLOBAL_LOAD_ASYNC_TO_LDS_B128 + s_wait_asynccnt).
// ---------------------------------------------------------------------------
__global__ void gemm_n64_wmma(const float* __restrict__ A,
                              const float* __restrict__ B,
                              float* __restrict__ Out, int N) {
  __shared__ float Bs[64 * 64];

  // Async-stage B (16 KB) into LDS: each thread copies 4x 16B chunks.
  {
    unsigned lds0 = (unsigned)(size_t)(void*)&Bs[0];
    for (int i = threadIdx.x; i < 1024; i += blockDim.x) {   // 1024 float4s
      unsigned ldsoff = lds0 + (unsigned)i * 16u;
      const float* gp = B + (size_t)i * 4;
      asm volatile("global_load_async_to_lds_b128 %0, %1, off"
                   :: "v"(ldsoff), "v"(gp) : "memory");
    }
    asm volatile("s_wait_asynccnt 0" ::: "memory");
  }
  __syncthreads();

  int gwave = (int)((blockIdx.x * blockDim.x + threadIdx.x) >> 5);
  int lane  = threadIdx.x & 31;
  int row0  = gwave * 16;
  if (row0 >= N) return;                 // wave-uniform: EXEC stays all-ones

  int m  = lane & 15;                    // A: M index / B,C,D: N index
  int kh = lane >> 4;                    // half-wave -> K pair select
  int rowA = row0 + m; if (rowA >= N) rowA = N - 1;
  const float* Arow = A + (size_t)rowA * 64;

  // Register-blocked A fragments: one 16x64 tile per wave, loaded once.
  v2f afrag[16];
#pragma unroll
  for (int t = 0; t < 16; ++t) {
    int ka = 4 * t + kh * 2;
    afrag[t].x = Arow[ka];
    afrag[t].y = Arow[ka + 1];
  }

  for (int nt = 0; nt < 4; ++nt) {
    v8f acc = {};
    int ncol = nt * 16 + m;
#pragma unroll
    for (int t = 0; t < 16; ++t) {
      int ka = 4 * t + kh * 2;
      v2f b;
      b.x = Bs[ka * 64 + ncol];
      b.y = Bs[(ka + 1) * 64 + ncol];
      acc = __builtin_amdgcn_wmma_f32_16x16x4_f32(false, afrag[t], false, b,
                                                  (short)0, acc, false, false);
    }
#pragma unroll
    for (int v = 0; v < 8; ++v) {        // C/D: VGPR v -> row (v + kh*8)
      int rr = row0 + kh * 8 + v;
      if (rr < N) Out[(size_t)rr * 64 + nt * 16 + m] = acc[v];
    }
  }
}

// ---------------------------------------------------------------------------
// Out[N x K] = A[N x 64] * W2[64 x K] + b2, K <= 64 (zero-padded in LDS,
// so B staging uses regular loads here).
// ---------------------------------------------------------------------------
__global__ void gemm_final_wmma(const float* __restrict__ A,
                                const float* __restrict__ W2,
                                const float* __restrict__ b2,
                                float* __restrict__ Out, int N, int K) {
  __shared__ float Bs[64 * 64];
  for (int i = threadIdx.x; i < 64 * 64; i += blockDim.x) {
    int r = i >> 6, c = i & 63;
    Bs[i] = (c < K) ? W2[r * K + c] : 0.0f;
  }
  __syncthreads();

  int gwave = (int)((blockIdx.x * blockDim.x + threadIdx.x) >> 5);
  int lane  = threadIdx.x & 31;
  int row0  = gwave * 16;
  if (row0 >= N) return;

  int m  = lane & 15;
  int kh = lane >> 4;
  int rowA = row0 + m; if (rowA >= N) rowA = N - 1;
  const float* Arow = A + (size_t)rowA * 64;

  v2f afrag[16];
#pragma unroll
  for (int t = 0; t < 16; ++t) {
    int ka = 4 * t + kh * 2;
    afrag[t].x = Arow[ka];
    afrag[t].y = Arow[ka + 1];
  }

  for (int nt = 0; nt < 4; ++nt) {
    int ncol = nt * 16 + m;
    v8f acc = {};
#pragma unroll
    for (int t = 0; t < 16; ++t) {
      int ka = 4 * t + kh * 2;
      v2f b;
      b.x = Bs[ka * 64 + ncol];
      b.y = Bs[(ka + 1) * 64 + ncol];
      acc = __builtin_amdgcn_wmma_f32_16x16x4_f32(false, afrag[t], false, b,
                                                  (short)0, acc, false, false);
    }
    if (ncol < K) {
      float bias = b2[ncol];
#pragma unroll
      for (int v = 0; v < 8; ++v) {
        int rr = row0 + kh * 8 + v;
        if (rr < N) Out[(size_t)rr * K + ncol] = acc[v] + bias;
      }
    }
  }
}

// GTConv filter: softmax over gt_w[l, c, :]  (R tiny) -> filt[0..R)
__global__ void filt_kernel(const float* __restrict__ gt_w, int l, int c,
                            int C, int R, float* __restrict__ filt) {
  const float* w = gt_w + (size_t)(l * C + c) * R;
  float mx = w[0];
  for (int r = 1; r < R; ++r) mx = fmaxf(mx, w[r]);
  float s = 0.0f;
  for (int r = 0; r < R; ++r) { float e = __expf(w[r] - mx); filt[r] = e; s += e; }
  for (int r = 0; r < R; ++r) filt[r] /= s;
}

// weighted in-degree: deg[dst] += filt[rel]
__global__ void deg_kernel(const int* __restrict__ rel, const int* __restrict__ dst,
                           const float* __restrict__ filt, float* __restrict__ deg, int E) {
  int e = blockIdx.x * blockDim.x + threadIdx.x;
  if (e >= E) return;
  atomicAdd(&deg[dst[e]], filt[rel[e]]);
}

// el/er = ft @ attn_l/attn_r (per node)
__global__ void elr_kernel(const float* __restrict__ ft, const float* __restrict__ al,
                           const float* __restrict__ ar, float* __restrict__ el,
                           float* __restrict__ er, int N) {
  int i = blockIdx.x * blockDim.x + threadIdx.x;
  if (i >= N) return;
  const float* f = ft + (size_t)i * 64;
  float sl = 0.0f, sr = 0.0f;
#pragma unroll
  for (int j = 0; j < 64; ++j) { float x = f[j]; sl += x * al[j]; sr += x * ar[j]; }
  el[i] = sl; er[i] = sr;
}

// logits = leaky_relu(el[src]+er[dst]); segment max into emax[dst]
__global__ void logits_kernel(const int* __restrict__ src, const int* __restrict__ dst,
                              const float* __restrict__ el, const float* __restrict__ er,
                              float* __restrict__ elog, float* __restrict__ emax, int E) {
  int e = blockIdx.x * blockDim.x + threadIdx.x;
  if (e >= E) return;
  float x = el[src[e]] + er[dst[e]];
  x = x > 0.0f ? x : LEAKY * x;
  elog[e] = x;
  atomicMaxF(&emax[dst[e]], x);
}

// a = exp(logit - emax[dst]) (in place); segment sum into esum[dst]
__global__ void expsum_kernel(const int* __restrict__ dst, float* __restrict__ elog,
                              const float* __restrict__ emax, float* __restrict__ esum, int E) {
  int e = blockIdx.x * blockDim.x + threadIdx.x;
  if (e >= E) return;
  int d = dst[e];
  float a = __expf(elog[e] - emax[d]);
  elog[e] = a;
  atomicAdd(&esum[d], a);
}

// pass 1: Y[dst] += X[src] * (filt[rel] / deg[dst])   (edge x float4 threads)
__global__ void spmm_ew_kernel(const int* __restrict__ src, const int* __restrict__ dst,
                               const int* __restrict__ rel, const float* __restrict__ filt,
                               const float* __restrict__ deg, const float* __restrict__ X,
                               float* __restrict__ Y, int E) {
  int t = blockIdx.x * blockDim.x + threadIdx.x;
  int e = t >> 4;
  if (e >= E) return;
  int q = t & 15;
  int d = dst[e];
  float dg = deg[d];
  float w = filt[rel[e]] / (dg > 0.0f ? dg : 1.0f);
  const float4 v = *(const float4*)(X + (size_t)src[e] * 64 + q * 4);
  float* o = Y + (size_t)d * 64 + q * 4;
  atomicAdd(o + 0, v.x * w);
  atomicAdd(o + 1, v.y * w);
  atomicAdd(o + 2, v.z * w);
  atomicAdd(o + 3, v.w * w);
}

// pass 2: Y[dst] += X[src] * (a[e] / esum[dst])
__global__ void spmm_att_kernel(const int* __restrict__ src, const int* __restrict__ dst,
                                const float* __restrict__ a, const float* __restrict__ esum,
                                const float* __restrict__ X, float* __restrict__ Y, int E) {
  int t = blockIdx.x * blockDim.x + threadIdx.x;
  int e = t >> 4;
  if (e >= E) return;
  int q = t & 15;
  int d = dst[e];
  float alpha = a[e] / esum[d];
  const float4 v = *(const float4*)(X + (size_t)src[e] * 64 + q * 4);
  float* o = Y + (size_t)d * 64 + q * 4;
  atomicAdd(o + 0, v.x * alpha);
  atomicAdd(o + 1, v.y * alpha);
  atomicAdd(o + 2, v.z * alpha);
  atomicAdd(o + 3, v.w * alpha);
}

// Hm += lin_w[c] * X
__global__ void axpy_kernel(const float* __restrict__ lin_w, int c,
                            const float* __restrict__ x, float* __restrict__ y, long n) {
  long i = (long)blockIdx.x * blockDim.x + threadIdx.x;
  if (i >= n) return;
  y[i] += lin_w[c] * x[i];
}

extern "C" void kernel_launch(void* const* d_in, const int* in_sizes, int n_in,
                              void* d_out, int out_size, void* d_ws, size_t ws_size,
                              hipStream_t stream) {
  const float* h      = (const float*)d_in[0];
  const float* params = (const float*)d_in[1];
  const float* gt_w   = (const float*)d_in[2];
  const float* fc_w   = (const float*)d_in[3];
  const float* attn_l = (const float*)d_in[4];
  const float* attn_r = (const float*)d_in[5];
  const float* lin_w  = (const float*)d_in[6];
  const float* W2     = (const float*)d_in[7];
  const float* b2     = (const float*)d_in[8];
  const int*   src    = (const int*)d_in[9];
  const int*   dst    = (const int*)d_in[10];
  const int*   rel    = (const int*)d_in[11];
  float* out = (float*)d_out;

  const int N = in_sizes[0] / 64;      // h is [N, 64]
  const int C = in_sizes[6];           // lin_w is [C]
  const int L = 2;                     // num_layers (fixed in reference)
  const int R = in_sizes[2] / (L * C); // gt_w is [L, C, R]
  const int E = in_sizes[9];           // src is [Etot]
  const int K = in_sizes[8];           // b2 is [K]
  const long NH = (long)N * 64;

  // workspace carve-up (all f32)
  float* ws   = (float*)d_ws;
  float* Hc   = ws; ws += NH;
  float* ft   = ws; ws += NH;
  float* ft1  = ws; ws += NH;
  float* ft2  = ws; ws += NH;
  float* Hm   = ws; ws += NH;
  float* deg  = ws; ws += N;
  float* el   = ws; ws += N;
  float* er   = ws; ws += N;
  float* emax = ws; ws += N;
  float* esum = ws; ws += N;
  float* elog = ws; ws += E;
  float* filt = ws; ws += 8;

  const int nwaves  = (N + 15) / 16;
  const int gblocks = (nwaves + 7) / 8;        // 8 waves per 256-thread block
  const int eblocks = (E + 255) / 256;
  const int sblocks = (E * 16 + 255) / 256;    // edge x float4-chunk
  const int nblocks = (N + 255) / 256;

  auto fill = [&](float* p, float v, long n) {
    fill_kernel<<<(int)((n + 255) / 256), 256, 0, stream>>>(p, v, n);
  };

  fill(Hm, 0.0f, NH);

  float* bHc  = Hc;
  float* bft2 = ft2;
  for (int c = 0; c < C; ++c) {
    // Hc = h @ params[c]
    gemm_n64_wmma<<<gblocks, 256, 0, stream>>>(h, params + (size_t)c * 64 * 64, bHc, N);
    for (int l = 0; l < L; ++l) {
      // GTConv: filt = softmax(gt_w[l])[c], edge-weight degree normalization
      filt_kernel<<<1, 1, 0, stream>>>(gt_w, l, c, C, R, filt);
      fill(deg, 0.0f, N);
      deg_kernel<<<eblocks, 256, 0, stream>>>(rel, dst, filt, deg, E);
      // GAT fc projection
      gemm_n64_wmma<<<gblocks, 256, 0, stream>>>(bHc, fc_w, ft, N);
      // attention logits + edge softmax
      elr_kernel<<<nblocks, 256, 0, stream>>>(ft, attn_l, attn_r, el, er, N);
      fill(emax, -INFINITY, N);
      logits_kernel<<<eblocks, 256, 0, stream>>>(src, dst, el, er, elog, emax, E);
      fill(esum, 0.0f, N);
      expsum_kernel<<<eblocks, 256, 0, stream>>>(dst, elog, emax, esum, E);
      // pass 1: edge-weight SpMM; pass 2: attention SpMM
      fill(ft1, 0.0f, NH);
      spmm_ew_kernel<<<sblocks, 256, 0, stream>>>(src, dst, rel, filt, deg, ft, ft1, E);
      fill(bft2, 0.0f, NH);
      spmm_att_kernel<<<sblocks, 256, 0, stream>>>(src, dst, elog, esum, ft1, bft2, E);
      float* tmp = bHc; bHc = bft2; bft2 = tmp;   // Hc <- ft2
    }
    // Hm += lin_w[c] * Hc
    axpy_kernel<<<(int)((NH + 255) / 256), 256, 0, stream>>>(lin_w, c, bHc, Hm, NH);
  }

  // logits = Hm @ W2 + b2
  gemm_final_wmma<<<gblocks, 256, 0, stream>>>(Hm, W2, b2, out, N, K);
}